// VerticalCompression_44624710205817
// MI455X (gfx1250) — compile-verified
//
#include <hip/hip_runtime.h>

// ---------------------------------------------------------------------------
// VerticalCompression on gfx1250 (MI455X), dense fp32 WMMA formulation.
//   out[b,c,o,p] = sum_{i,w} comp[c,o,i] * inp'[b,i,w] * comp[c,p,w]
//                  / (rs[c,o]*rs[c,p]),  comp = mask*weight, inp' = diag-zeroed
// Two tiled GEMMs via V_WMMA_F32_16X16X4_F32, tmp staged in d_ws (128 MB).
// GEMM1 B tile stored k-pair-interleaved in LDS so every WMMA operand is one
// aligned b64 LDS load. GEMM2 A tile staged via async global->LDS DMA
// (GLOBAL_LOAD_ASYNC_TO_LDS_B128). global_prefetch_b8 used for next K-tile.
// ---------------------------------------------------------------------------

typedef float v2f __attribute__((ext_vector_type(2)));
typedef float v8f __attribute__((ext_vector_type(8)));
typedef int   v4i_vs __attribute__((vector_size(16)));   // matches builtin proto
typedef __attribute__((address_space(1))) v4i_vs* async_gptr;
typedef __attribute__((address_space(3))) v4i_vs* async_lptr;

#define B_SZ    4
#define C_SZ    4
#define N_IN    8192
#define N_OUT   1024
#define M1      (C_SZ * N_OUT)       // 4096 rows of GEMM1 (c,o)
#define KT      32                   // K tile
#define MT      128                  // M tile
#define NT      128                  // N tile
#define LDA     36                   // padded LDS pitch for [row][k] tiles
#define PN      144                  // interleaved pitch: 2*PN mod 64 == 32

#if defined(__has_builtin)
#  if __has_builtin(__builtin_amdgcn_global_load_async_to_lds_b128) && \
      __has_builtin(__builtin_amdgcn_s_wait_asynccnt)
#    define HAVE_ASYNC_LDS 1
#  endif
#endif
#ifndef HAVE_ASYNC_LDS
#  define HAVE_ASYNC_LDS 0
#endif

// ----------------------------- rs = comp.sum(-1) ---------------------------
__global__ __launch_bounds__(256)
void rs_kernel(const float* __restrict__ mask, const float* __restrict__ weight,
               float* __restrict__ rs) {
    const int row  = blockIdx.x * 8 + (threadIdx.x >> 5);   // 0..4095  (c*1024+o)
    const int lane = threadIdx.x & 31;
    const size_t mrow = (size_t)(row & (N_OUT - 1)) * N_IN;
    const size_t wrow = (size_t)row * N_IN;
    float s = 0.f;
    for (int k = lane; k < N_IN; k += 32)
        s += mask[mrow + k] * weight[wrow + k];
    #pragma unroll
    for (int off = 16; off > 0; off >>= 1)
        s += __shfl_down(s, off, 32);
    if (lane == 0) rs[row] = s;
}

// --------------------- GEMM1: tmp[m][w] = comp[m][:] * inp'[b] -------------
__global__ __launch_bounds__(256)
void gemm1_kernel(const float* __restrict__ inpb,   // input + b*8192*8192
                  const float* __restrict__ mask,
                  const float* __restrict__ weight,
                  float* __restrict__ tmp) {
    __shared__ float A_lds[MT * LDA];            // [m][k] comp tile    (18 KB)
    __shared__ float B2_lds[(KT / 2) * PN * 2];  // k-pair interleaved  (18 KB)

    const int tid  = threadIdx.x;
    const int lane = tid & 31;
    const int lr   = lane & 15;         // matrix row/col within 16
    const int lh   = lane >> 4;         // 0/1 half-wave
    const int wave = tid >> 5;
    const int wm   = (wave >> 1) * 32;  // wave M offset in tile
    const int wn   = (wave & 1) * 64;   // wave N offset in tile

    const int m0 = blockIdx.y * MT;     // row block  (c*1024+o)
    const int n0 = blockIdx.x * NT;     // col block  (w)

    v8f acc[2][4];
    const v8f vzero = {0.f, 0.f, 0.f, 0.f, 0.f, 0.f, 0.f, 0.f};
    #pragma unroll
    for (int im = 0; im < 2; ++im)
        #pragma unroll
        for (int in = 0; in < 4; ++in) acc[im][in] = vzero;

    const int arow = tid >> 3;          // 0..31
    const int acol = (tid & 7) * 4;     // 0..28
    const int bn   = tid & 127;         // n within tile
    const int bkh  = tid >> 7;          // 0/1: k-pair parity group

    for (int kt = 0; kt < N_IN; kt += KT) {
        if (kt + KT < N_IN) {   // pull next K-tile toward GL2
            __builtin_prefetch(weight + (size_t)(m0 + arow) * N_IN + kt + KT + acol, 0, 1);
            __builtin_prefetch(inpb + (size_t)(kt + KT + 2 * bkh) * N_IN + n0 + bn, 0, 1);
        }
        // -- stage A tile: comp = mask*weight, rows m0..m0+127, cols kt..kt+31
        #pragma unroll
        for (int pass = 0; pass < 4; ++pass) {
            const int r = arow + pass * 32;
            const int m = m0 + r;
            const float4 mk = *(const float4*)(mask   + (size_t)(m & (N_OUT-1)) * N_IN + kt + acol);
            const float4 wt = *(const float4*)(weight + (size_t)m * N_IN + kt + acol);
            float4 v;
            v.x = mk.x * wt.x; v.y = mk.y * wt.y; v.z = mk.z * wt.z; v.w = mk.w * wt.w;
            *(float4*)&A_lds[r * LDA + acol] = v;
        }
        // -- stage B tile, k-pair interleaved: B2[(kp*PN+n)*2+{0,1}] = inp'[2kp(+1)][n]
        #pragma unroll
        for (int pp = 0; pp < 8; ++pp) {
            const int kp = pp * 2 + bkh;            // pair 0..15
            const int i0 = kt + 2 * kp;
            const int wg = n0 + bn;
            v2f v;
            v.x = inpb[(size_t)i0 * N_IN + wg];
            v.y = inpb[(size_t)(i0 + 1) * N_IN + wg];
            if (i0     == wg) v.x = 0.f;            // diag removal
            if (i0 + 1 == wg) v.y = 0.f;
            *(v2f*)&B2_lds[(kp * PN + bn) * 2] = v;
        }
        __syncthreads();

        #pragma unroll
        for (int kk = 0; kk < KT; kk += 4) {
            const int kc = kk + 2 * lh;             // lane's K pair base
            const int kp = kc >> 1;
            v2f afrag[2], bfrag[4];
            #pragma unroll
            for (int im = 0; im < 2; ++im)
                afrag[im] = *(const v2f*)&A_lds[(wm + im * 16 + lr) * LDA + kc];
            #pragma unroll
            for (int in = 0; in < 4; ++in)
                bfrag[in] = *(const v2f*)&B2_lds[(kp * PN + wn + in * 16 + lr) * 2];
            #pragma unroll
            for (int im = 0; im < 2; ++im)
                #pragma unroll
                for (int in = 0; in < 4; ++in)
                    acc[im][in] = __builtin_amdgcn_wmma_f32_16x16x4_f32(
                        false, afrag[im], false, bfrag[in],
                        (short)0, acc[im][in], false, false);
        }
        __syncthreads();
    }

    // -- write tmp
    #pragma unroll
    for (int im = 0; im < 2; ++im)
        #pragma unroll
        for (int in = 0; in < 4; ++in)
            #pragma unroll
            for (int j = 0; j < 8; ++j) {
                const int row = m0 + wm + im * 16 + lh * 8 + j;
                const int col = n0 + wn + in * 16 + lr;
                tmp[(size_t)row * N_IN + col] = acc[im][in][j];
            }
}

// ------------- GEMM2: out[b,c,o,p] = tmp[c,o,:]·comp[c,p,:] / norm ---------
__global__ __launch_bounds__(256)
void gemm2_kernel(const float* __restrict__ tmp,
                  const float* __restrict__ mask,
                  const float* __restrict__ weight,
                  const float* __restrict__ rs,
                  float* __restrict__ outb) {        // out + b*C*1024*1024
    __shared__ float A_lds[MT * LDA];   // [o][w] tmp tile
    __shared__ float Bt_lds[NT * LDA];  // [p][w] comp tile (row-major in K)

    const int tid  = threadIdx.x;
    const int lane = tid & 31;
    const int lr   = lane & 15;
    const int lh   = lane >> 4;
    const int wave = tid >> 5;
    const int wm   = (wave >> 1) * 32;
    const int wn   = (wave & 1) * 64;

    const int c  = blockIdx.z;
    const int m0 = blockIdx.y * MT;     // o block
    const int n0 = blockIdx.x * NT;     // p block

    v8f acc[2][4];
    const v8f vzero = {0.f, 0.f, 0.f, 0.f, 0.f, 0.f, 0.f, 0.f};
    #pragma unroll
    for (int im = 0; im < 2; ++im)
        #pragma unroll
        for (int in = 0; in < 4; ++in) acc[im][in] = vzero;

    const int arow = tid >> 3;
    const int acol = (tid & 7) * 4;

    for (int kt = 0; kt < N_IN; kt += KT) {
        if (kt + KT < N_IN) {
            __builtin_prefetch(tmp + (size_t)(c * N_OUT + m0 + arow) * N_IN + kt + KT + acol, 0, 1);
            __builtin_prefetch(weight + (size_t)(c * N_OUT + n0 + arow) * N_IN + kt + KT + acol, 0, 1);
        }
        #pragma unroll
        for (int pass = 0; pass < 4; ++pass) {
            const int r = arow + pass * 32;
            // A tile: tmp rows (c*1024 + m0 + r)  -- pure copy, async DMA if available
#if HAVE_ASYNC_LDS
            __builtin_amdgcn_global_load_async_to_lds_b128(
                (async_gptr)(tmp + (size_t)(c * N_OUT + m0 + r) * N_IN + kt + acol),
                (async_lptr)&A_lds[r * LDA + acol],
                0, 0);
#else
            const float4 av = *(const float4*)(tmp + (size_t)(c * N_OUT + m0 + r) * N_IN + kt + acol);
            *(float4*)&A_lds[r * LDA + acol] = av;
#endif
            // B tile: comp rows (p = n0 + r), stored [p][w]
            const int p = n0 + r;
            const float4 mk = *(const float4*)(mask   + (size_t)p * N_IN + kt + acol);
            const float4 wt = *(const float4*)(weight + (size_t)(c * N_OUT + p) * N_IN + kt + acol);
            float4 bv;
            bv.x = mk.x * wt.x; bv.y = mk.y * wt.y; bv.z = mk.z * wt.z; bv.w = mk.w * wt.w;
            *(float4*)&Bt_lds[r * LDA + acol] = bv;
        }
#if HAVE_ASYNC_LDS
        __builtin_amdgcn_s_wait_asynccnt(0);
#endif
        __syncthreads();

        #pragma unroll
        for (int kk = 0; kk < KT; kk += 4) {
            const int kc = kk + 2 * lh;
            v2f afrag[2], bfrag[4];
            #pragma unroll
            for (int im = 0; im < 2; ++im)
                afrag[im] = *(const v2f*)&A_lds[(wm + im * 16 + lr) * LDA + kc];
            #pragma unroll
            for (int in = 0; in < 4; ++in)
                bfrag[in] = *(const v2f*)&Bt_lds[(wn + in * 16 + lr) * LDA + kc];
            #pragma unroll
            for (int im = 0; im < 2; ++im)
                #pragma unroll
                for (int in = 0; in < 4; ++in)
                    acc[im][in] = __builtin_amdgcn_wmma_f32_16x16x4_f32(
                        false, afrag[im], false, bfrag[in],
                        (short)0, acc[im][in], false, false);
        }
        __syncthreads();
    }

    // -- renormalize + store folded output (b*C + c, o, p)
    #pragma unroll
    for (int in = 0; in < 4; ++in) {
        const int p  = n0 + wn + in * 16 + lr;
        const float rp = rs[c * N_OUT + p];
        #pragma unroll
        for (int im = 0; im < 2; ++im)
            #pragma unroll
            for (int j = 0; j < 8; ++j) {
                const int o = m0 + wm + im * 16 + lh * 8 + j;
                float norm = rs[c * N_OUT + o] * rp;
                norm = (norm == 0.f) ? 1.f : norm;
                outb[(size_t)c * N_OUT * N_OUT + (size_t)o * N_OUT + p] =
                    acc[im][in][j] / norm;
            }
    }
}

// ---------------------------------------------------------------------------
extern "C" void kernel_launch(void* const* d_in, const int* in_sizes, int n_in,
                              void* d_out, int out_size, void* d_ws, size_t ws_size,
                              hipStream_t stream) {
    const float* input  = (const float*)d_in[0];   // (B, 8192, 8192)
    const float* mask   = (const float*)d_in[1];   // (1024, 8192)
    const float* weight = (const float*)d_in[2];   // (4, 1024, 8192)
    float*       out    = (float*)d_out;           // (16, 1024, 1024)

    float* tmp = (float*)d_ws;                                        // 128 MB
    float* rs  = (float*)((char*)d_ws + (size_t)M1 * N_IN * sizeof(float));

    rs_kernel<<<M1 / 8, 256, 0, stream>>>(mask, weight, rs);

    for (int b = 0; b < B_SZ; ++b) {
        const float* inpb = input + (size_t)b * N_IN * N_IN;
        gemm1_kernel<<<dim3(N_IN / NT, M1 / MT), 256, 0, stream>>>(
            inpb, mask, weight, tmp);
        gemm2_kernel<<<dim3(N_OUT / NT, N_OUT / MT, C_SZ), 256, 0, stream>>>(
            tmp, mask, weight, rs,
            out + (size_t)b * C_SZ * N_OUT * N_OUT);
    }
    (void)in_sizes; (void)n_in; (void)out_size; (void)ws_size;
}